// CRFLayer_36524401885736
// MI455X (gfx1250) — compile-verified
//
#include <hip/hip_runtime.h>
#include <cstdint>

#define B_      512
#define S_      4096
#define T_      8
#define CHUNKS  32
#define CLEN    128          // S_/CHUNKS
#define NEG_INF (-1.0e30f)
#define LOG2E   1.44269504088896340736f
#define LN2     0.69314718055994530942f

typedef float v2f __attribute__((ext_vector_type(2)));
typedef float v8f __attribute__((ext_vector_type(8)));

// Raw hardware exp2/log2 (no denorm fixup; all our inputs are safely normal).
__device__ __forceinline__ float fast_exp(float x) {
    return __builtin_amdgcn_exp2f(x * LOG2E);
}
__device__ __forceinline__ float fast_log(float x) {
    return __builtin_amdgcn_logf(x) * LN2;
}

// ---------------------------------------------------------------------------
// K1: chunk scan. thread = (batch b, chunk c, matrix row).
// Propagates row `row` of the 8x8 log-semiring transfer matrix for chunk c:
//   R <- R (x) M_s,  M_s[k][j] = trans[k][j] + em[s][j]
// via R_new[j] = em[j] + m + log( sum_k exp(R[k]-m) * E[k][j] ), E = exp(trans).
// Row-0 lanes additionally accumulate the gold path score + mask count.
// ---------------------------------------------------------------------------
__global__ __launch_bounds__(256) void crf_chunk_scan(
    const float* __restrict__ em,       // [B,S,T]
    const float* __restrict__ trans,    // [T,T]
    const int* __restrict__ tags,       // [B,S]
    const uint8_t* __restrict__ mask,   // [B,S] (numpy bool, 1 byte)
    float* __restrict__ G,              // [B,CHUNKS,8,8]
    float* __restrict__ goldPart,       // [B,CHUNKS]
    float* __restrict__ maskPart)       // [B,CHUNKS]
{
    const int tid = blockIdx.x * 256 + threadIdx.x;
    const int row = tid & 7;
    const int c   = (tid >> 3) & (CHUNKS - 1);
    const int b   = tid >> 8;           // tid / (8*CHUNKS)

    __shared__ float sTr[64];
    if (threadIdx.x < 64) sTr[threadIdx.x] = trans[threadIdx.x];
    __syncthreads();

    float E[8][8];
    #pragma unroll
    for (int k = 0; k < 8; ++k)
        #pragma unroll
        for (int j = 0; j < 8; ++j)
            E[k][j] = fast_exp(trans[k * 8 + j]);

    float R[8];
    #pragma unroll
    for (int k = 0; k < 8; ++k) R[k] = (k == row) ? 0.0f : NEG_INF;

    const float*   emb = em   + (size_t)b * S_ * T_;
    const int*     tgb = tags + (size_t)b * S_;
    const uint8_t* mkb = mask + (size_t)b * S_;

    const int s0 = c * CLEN + (c == 0 ? 1 : 0);   // s=0 is alpha0, no matrix
    const int s1 = (c + 1) * CLEN;

    float gold = 0.0f, mcnt = 0.0f;
    int pt = (row == 0) ? tgb[s0 - 1] : 0;

    for (int s = s0; s < s1; ++s) {
        const float4 e0 = *(const float4*)(emb + (size_t)s * 8);
        const float4 e1 = *(const float4*)(emb + (size_t)s * 8 + 4);
        const uint8_t mk = mkb[s];
        if (mk) {
            float m = R[0];
            #pragma unroll
            for (int k = 1; k < 8; ++k) m = fmaxf(m, R[k]);
            const float mm = -m * LOG2E;
            float acc[8] = {0, 0, 0, 0, 0, 0, 0, 0};
            #pragma unroll
            for (int k = 0; k < 8; ++k) {
                const float pk = __builtin_amdgcn_exp2f(fmaf(R[k], LOG2E, mm));
                #pragma unroll
                for (int j = 0; j < 8; ++j) acc[j] = fmaf(pk, E[k][j], acc[j]);
            }
            R[0] = fmaf(__builtin_amdgcn_logf(acc[0]), LN2, e0.x + m);
            R[1] = fmaf(__builtin_amdgcn_logf(acc[1]), LN2, e0.y + m);
            R[2] = fmaf(__builtin_amdgcn_logf(acc[2]), LN2, e0.z + m);
            R[3] = fmaf(__builtin_amdgcn_logf(acc[3]), LN2, e0.w + m);
            R[4] = fmaf(__builtin_amdgcn_logf(acc[4]), LN2, e1.x + m);
            R[5] = fmaf(__builtin_amdgcn_logf(acc[5]), LN2, e1.y + m);
            R[6] = fmaf(__builtin_amdgcn_logf(acc[6]), LN2, e1.z + m);
            R[7] = fmaf(__builtin_amdgcn_logf(acc[7]), LN2, e1.w + m);
        }
        if (row == 0) {
            const int t = tgb[s];
            if (mk) {
                gold += sTr[pt * 8 + t] + emb[(size_t)s * 8 + t];
                mcnt += 1.0f;
            }
            pt = t;
        }
    }

    float* g = G + (size_t)(b * CHUNKS + c) * 64 + row * 8;
    #pragma unroll
    for (int j = 0; j < 8; ++j) g[j] = R[j];
    if (row == 0) {
        goldPart[b * CHUNKS + c] = gold;
        maskPart[b * CHUNKS + c] = mcnt;
    }
}

// ---------------------------------------------------------------------------
// K2: combine chunk matrices pairwise in the log-semiring via WMMA.
// (A (x) B)[i][j] = mA[i] + mB[j] + log( sum_k e^{A[i][k]-mA[i]} e^{B[k][j]-mB[j]} )
// One wave handles TWO independent 8x8 combines packed block-diagonally into a
// 16x16 f32 matmul = 4x V_WMMA_F32_16X16X4_F32 (K=16), operands staged in LDS.
// A layout (ISA 7.12.2): lanes 0-15 M=lane K={kk,kk+1}; lanes 16-31 K={kk+2,kk+3}.
// B layout assumed transposed-symmetric (K split across lane halves, N=lane%16).
// ---------------------------------------------------------------------------
__global__ __launch_bounds__(256) void crf_combine_wmma(
    float* __restrict__ G, int level, int pairsPerBatch)
{
    __shared__ float sPA[8][16][16];
    __shared__ float sPB[8][16][16];
    __shared__ float sMA[8][16];
    __shared__ float sMB[8][16];

    const int w    = threadIdx.x >> 5;
    const int lane = threadIdx.x & 31;
    const int half = lane >> 4;
    const int l16  = lane & 15;

    const int waveId = blockIdx.x * 8 + w;
    const int stride = 1 << level;

    const int p0 = waveId * 2;           // grid sized so p0 and p0+1 are valid
    const int b0 = p0 / pairsPerBatch,     i0 = p0 % pairsPerBatch;
    const int b1 = (p0+1) / pairsPerBatch, i1 = (p0+1) % pairsPerBatch;
    const int left0 = i0 * (stride * 2), right0 = left0 + stride;
    const int left1 = i1 * (stride * 2), right1 = left1 + stride;

    float* out0 = G + (size_t)(b0 * CHUNKS + left0) * 64;
    float* out1 = G + (size_t)(b1 * CHUNKS + left1) * 64;
    const float* ra = (l16 < 8) ? out0 : out1;                                  // A source
    const float* rb = (l16 < 8) ? (G + (size_t)(b0 * CHUNKS + right0) * 64)
                                : (G + (size_t)(b1 * CHUNKS + right1) * 64);    // B source

    // Phase 1: zero one 16-wide row each, then fill raw 8x8 blocks (same lane).
    const int r  = l16;
    const int co = (r < 8) ? 0 : 8;
    const int rr = r & 7;
    if (half == 0) {
        #pragma unroll
        for (int k = 0; k < 16; ++k) sPA[w][r][k] = 0.0f;
        #pragma unroll
        for (int k = 0; k < 8; ++k)  sPA[w][r][co + k] = ra[rr * 8 + k];
    } else {
        #pragma unroll
        for (int k = 0; k < 16; ++k) sPB[w][r][k] = 0.0f;
        #pragma unroll
        for (int k = 0; k < 8; ++k)  sPB[w][r][co + k] = rb[rr * 8 + k];
    }
    __syncthreads();

    // Phase 2: row-max of A rows, col-max of B cols; overwrite with exp.
    if (half == 0) {
        float m = sPA[w][r][co];
        #pragma unroll
        for (int k = 1; k < 8; ++k) m = fmaxf(m, sPA[w][r][co + k]);
        sMA[w][r] = m;
        #pragma unroll
        for (int k = 0; k < 8; ++k)
            sPA[w][r][co + k] = fast_exp(sPA[w][r][co + k] - m);
    } else {
        const int j = l16, cb = (j < 8) ? 0 : 8;
        float m = sPB[w][cb][j];
        #pragma unroll
        for (int k = 1; k < 8; ++k) m = fmaxf(m, sPB[w][cb + k][j]);
        sMB[w][j] = m;
        #pragma unroll
        for (int k = 0; k < 8; ++k)
            sPB[w][cb + k][j] = fast_exp(sPB[w][cb + k][j] - m);
    }
    __syncthreads();

    // Phase 3: 16x16x16 f32 matmul via 4 accumulating WMMAs.
    v8f acc = {0, 0, 0, 0, 0, 0, 0, 0};
    #pragma unroll
    for (int kk = 0; kk < 16; kk += 4) {
        v2f a, bb;
        a.x  = sPA[w][l16][kk + 2 * half + 0];
        a.y  = sPA[w][l16][kk + 2 * half + 1];
        bb.x = sPB[w][kk + 2 * half + 0][l16];
        bb.y = sPB[w][kk + 2 * half + 1][l16];
        acc = __builtin_amdgcn_wmma_f32_16x16x4_f32(
                  false, a, false, bb, (short)0, acc, false, false);
    }

    // Phase 4: D layout: lane l, VGPR rD -> (M = rD + 8*half, N = l16).
    // Valid block-diagonal halves: (half==0, N<8) -> pair0; (half==1, N>=8) -> pair1.
    if (half == 0 && l16 < 8) {
        const float mb = sMB[w][l16];
        #pragma unroll
        for (int rd = 0; rd < 8; ++rd)
            out0[rd * 8 + l16] = sMA[w][rd] + mb + fast_log(acc[rd]);
    } else if (half == 1 && l16 >= 8) {
        const float mb = sMB[w][l16];
        const int jj = l16 - 8;
        #pragma unroll
        for (int rd = 0; rd < 8; ++rd)
            out1[rd * 8 + jj] = sMA[w][8 + rd] + mb + fast_log(acc[rd]);
    }
}

// ---------------------------------------------------------------------------
// K3: per-batch finalize (one block, thread b = batch item) + mean reduction.
// ---------------------------------------------------------------------------
__global__ __launch_bounds__(512) void crf_finalize(
    const float* __restrict__ em,
    const float* __restrict__ startT,
    const float* __restrict__ endT,
    const int* __restrict__ tags,
    const uint8_t* __restrict__ mask,
    const float* __restrict__ G,
    const float* __restrict__ goldPart,
    const float* __restrict__ maskPart,
    float* __restrict__ out)
{
    __shared__ float red[512];
    const int b = threadIdx.x;

    const float* Gf = G + (size_t)(b * CHUNKS) * 64;   // combined matrix at slot 0
    float alpha[8];
    #pragma unroll
    for (int j = 0; j < 8; ++j)
        alpha[j] = startT[j] + em[(size_t)b * S_ * 8 + j];

    float vmax = NEG_INF, v[8];
    #pragma unroll
    for (int j = 0; j < 8; ++j) {
        float m = alpha[0] + Gf[0 * 8 + j];
        #pragma unroll
        for (int k = 1; k < 8; ++k) m = fmaxf(m, alpha[k] + Gf[k * 8 + j]);
        float s = 0.0f;
        #pragma unroll
        for (int k = 0; k < 8; ++k) s += fast_exp(alpha[k] + Gf[k * 8 + j] - m);
        v[j] = m + fast_log(s) + endT[j];
        vmax = fmaxf(vmax, v[j]);
    }
    float se = 0.0f;
    #pragma unroll
    for (int j = 0; j < 8; ++j) se += fast_exp(v[j] - vmax);
    const float logZ = vmax + fast_log(se);

    float gold = 0.0f;
    float cnt  = (float)mask[(size_t)b * S_];          // mask[b,0] (s=0 not in chunks)
    #pragma unroll
    for (int c = 0; c < CHUNKS; ++c) {
        gold += goldPart[b * CHUNKS + c];
        cnt  += maskPart[b * CHUNKS + c];
    }
    const int t0 = tags[(size_t)b * S_];
    gold += startT[t0] + em[(size_t)b * S_ * 8 + t0];
    const int seqlen = (int)(cnt + 0.5f);
    const int lastT  = tags[(size_t)b * S_ + (seqlen - 1)];
    gold += endT[lastT];

    red[b] = logZ - gold;
    __syncthreads();
    for (int off = 256; off > 0; off >>= 1) {
        if (b < off) red[b] += red[b + off];
        __syncthreads();
    }
    if (b == 0) out[0] = red[0] * (1.0f / (float)B_);
}

// ---------------------------------------------------------------------------
extern "C" void kernel_launch(void* const* d_in, const int* in_sizes, int n_in,
                              void* d_out, int out_size, void* d_ws, size_t ws_size,
                              hipStream_t stream) {
    const float*   em     = (const float*)d_in[0];
    const float*   trans  = (const float*)d_in[1];
    const float*   startT = (const float*)d_in[2];
    const float*   endT   = (const float*)d_in[3];
    const int*     tags   = (const int*)d_in[4];
    const uint8_t* mask   = (const uint8_t*)d_in[5];

    float* ws       = (float*)d_ws;
    float* G        = ws;                              // 512*32*64 floats (4 MB)
    float* goldPart = G + (size_t)B_ * CHUNKS * 64;    // 16384 floats
    float* maskPart = goldPart + (size_t)B_ * CHUNKS;  // 16384 floats

    // Stage 1: 131072 threads (b, chunk, row) build chunk transfer matrices.
    crf_chunk_scan<<<(B_ * CHUNKS * 8) / 256, 256, 0, stream>>>(
        em, trans, tags, mask, G, goldPart, maskPart);

    // Stage 2: WMMA tree-combine 32 -> 1 matrices per batch (5 levels).
    for (int lvl = 0; lvl < 5; ++lvl) {
        const int ppb    = (CHUNKS / 2) >> lvl;   // pairs per batch: 16..1
        const int pairs  = B_ * ppb;              // 8192 .. 512
        const int blocks = pairs / 16;            // 2 pairs/wave, 8 waves/block
        crf_combine_wmma<<<blocks, 256, 0, stream>>>(G, lvl, ppb);
    }

    // Stage 3: finalize logZ, gold, nll mean.
    crf_finalize<<<1, 512, 0, stream>>>(em, startT, endT, tags, mask,
                                        G, goldPart, maskPart, (float*)d_out);
}